// SCAESuite_49091476193370
// MI455X (gfx1250) — compile-verified
//
#include <hip/hip_runtime.h>
#include <hip/hip_bf16.h>

typedef __attribute__((ext_vector_type(2))) float v2f;
typedef __attribute__((ext_vector_type(8))) float v8f;

#define F_  16384
#define D_  768
#define C_  16
#define BS_ 512

// ---------------------------------------------------------------------------
// Generic tiled transpose: in [R, Cc] -> out [Cc, R].  All dims here are
// multiples of 32 (768, 512, 16384) so no guards needed.
// ---------------------------------------------------------------------------
__global__ __launch_bounds__(256) void transpose_k(
    const float* __restrict__ in, float* __restrict__ out, int R, int Cc)
{
  __shared__ float tile[32][33];
  const int cb = blockIdx.x * 32;
  const int rb = blockIdx.y * 32;
  const int x  = threadIdx.x;
  #pragma unroll
  for (int yy = threadIdx.y; yy < 32; yy += 8)
    tile[yy][x] = in[(size_t)(rb + yy) * Cc + (cb + x)];
  __syncthreads();
  #pragma unroll
  for (int yy = threadIdx.y; yy < 32; yy += 8)
    out[(size_t)(cb + yy) * R + (rb + x)] = tile[x][yy];
}

// ---------------------------------------------------------------------------
// Stage 1: values[i,c] = E[i,:] . UT[j[i,c],:]  via f32 WMMA diagonal trick.
// One block = 16 consecutive i.  A = E[i0:i0+16, :] staged in LDS (reused by
// all 16 connection slots).  4 waves x 4 slots each; each wave keeps 4
// 16x16 f32 accumulators and extracts their diagonals at the end.
// ---------------------------------------------------------------------------
__global__ __launch_bounds__(128) void stage1_values(
    const float* __restrict__ E,    // down_encoder [F, D]
    const float* __restrict__ UT,   // up_decoder^T [F, D]
    const int*   __restrict__ jidx, // [F*C]
    float*       __restrict__ vals) // [F*C]
{
  __shared__ float As[16][D_ + 4];          // 16 x 772 floats = 48.25 KB
  const int tid  = threadIdx.x;
  const int lane = tid & 31;
  const int wave = tid >> 5;
  const int i0   = blockIdx.x * 16;

  // Cooperative, fully coalesced load of the 16 A rows.
  for (int idx = tid; idx < 16 * D_; idx += 128) {
    int r = idx / D_, k = idx - r * D_;
    As[r][k] = E[(size_t)(i0 + r) * D_ + k];
  }
  __syncthreads();

  const int m  = lane & 15;           // A row (M) == B col (N) for this lane
  const int kp = (lane >> 4) << 1;    // K sub-offset: 0 (lanes 0-15) / 2 (16-31)
  const int cbase = wave * 4;

  const float* Brow0 = UT + (size_t)jidx[(i0 + m) * C_ + cbase + 0] * D_;
  const float* Brow1 = UT + (size_t)jidx[(i0 + m) * C_ + cbase + 1] * D_;
  const float* Brow2 = UT + (size_t)jidx[(i0 + m) * C_ + cbase + 2] * D_;
  const float* Brow3 = UT + (size_t)jidx[(i0 + m) * C_ + cbase + 3] * D_;

  v8f acc0 = {}, acc1 = {}, acc2 = {}, acc3 = {};

  #pragma unroll 4
  for (int k0 = 0; k0 < D_; k0 += 4) {
    // A fragment: lane m, K = {k0+kp, k0+kp+1}   (one ds b64 read)
    v2f a  = *(const v2f*)&As[m][k0 + kp];
    // B fragments: gathered UT rows, K = {k0+kp, k0+kp+1}, N = m
    v2f b0 = *(const v2f*)&Brow0[k0 + kp];
    v2f b1 = *(const v2f*)&Brow1[k0 + kp];
    v2f b2 = *(const v2f*)&Brow2[k0 + kp];
    v2f b3 = *(const v2f*)&Brow3[k0 + kp];
    acc0 = __builtin_amdgcn_wmma_f32_16x16x4_f32(false, a, false, b0, (short)0, acc0, false, false);
    acc1 = __builtin_amdgcn_wmma_f32_16x16x4_f32(false, a, false, b1, (short)0, acc1, false, false);
    acc2 = __builtin_amdgcn_wmma_f32_16x16x4_f32(false, a, false, b2, (short)0, acc2, false, false);
    acc3 = __builtin_amdgcn_wmma_f32_16x16x4_f32(false, a, false, b3, (short)0, acc3, false, false);
  }

  // Diagonal extraction per the 16x16 f32 C/D layout:
  //  (r,r) r<8  -> lane r,    component r
  //  (r,r) r>=8 -> lane r+16, component r-8
  // Branch-free component select (cndmask chain; avoids scratch spill from
  // dynamic vector indexing).
  const bool lo  = (lane < 8);
  const bool hi  = (lane >= 16) && (lane < 24);
  const int comp = lo ? lane : (lane - 16);   // valid only when lo||hi
  const int row  = lo ? lane : (lane - 8);

  #pragma unroll
  for (int cc = 0; cc < 4; ++cc) {
    v8f a = (cc == 0) ? acc0 : (cc == 1) ? acc1 : (cc == 2) ? acc2 : acc3;
    float d = a[0];
    d = (comp == 1) ? a[1] : d;
    d = (comp == 2) ? a[2] : d;
    d = (comp == 3) ? a[3] : d;
    d = (comp == 4) ? a[4] : d;
    d = (comp == 5) ? a[5] : d;
    d = (comp == 6) ? a[6] : d;
    d = (comp == 7) ? a[7] : d;
    if (lo || hi)
      vals[(size_t)(i0 + row) * C_ + (cbase + cc)] = d;
  }
}

// ---------------------------------------------------------------------------
// Stage 2: outT[i, bs] = sum_c vals[i,c] * XT[j[i,c], bs].
// One block per feature i; all reads coalesced 2KB rows (L2 resident).
// ---------------------------------------------------------------------------
__global__ __launch_bounds__(256) void stage2_scatter(
    const float* __restrict__ XT,    // up_facts^T [F, BS]
    const float* __restrict__ vals,  // [F, C]
    const int*   __restrict__ jidx,  // [F*C]
    float* __restrict__ outT)        // [F, BS]
{
  __shared__ float vsh[C_];
  __shared__ int   jsh[C_];
  const int i   = blockIdx.x;
  const int tid = threadIdx.x;
  if (tid < C_) {
    vsh[tid] = vals[i * C_ + tid];
    jsh[tid] = jidx[i * C_ + tid];
  }
  __syncthreads();

  const float* rows[C_];
  #pragma unroll
  for (int c = 0; c < C_; ++c) {
    rows[c] = XT + (size_t)jsh[c] * BS_;
    __builtin_prefetch(rows[c] + tid, 0, 0);   // global_prefetch_b8
  }

  #pragma unroll
  for (int q = 0; q < 2; ++q) {
    const int bs = tid + q * 256;
    float acc = 0.f;
    #pragma unroll
    for (int c = 0; c < C_; ++c) acc += vsh[c] * rows[c][bs];
    outT[(size_t)i * BS_ + bs] = acc;
  }
}

// ---------------------------------------------------------------------------
extern "C" void kernel_launch(void* const* d_in, const int* in_sizes, int n_in,
                              void* d_out, int out_size, void* d_ws, size_t ws_size,
                              hipStream_t stream) {
  (void)in_sizes; (void)n_in; (void)out_size; (void)ws_size;
  const float* up_facts = (const float*)d_in[0];   // [B*S, F] = [512, 16384]
  const float* down_enc = (const float*)d_in[1];   // [F, D]
  const float* up_dec   = (const float*)d_in[2];   // [D, F]
  const int*   jidx     = (const int*)d_in[4];     // [F*C]  (i_indices = m/16 implicit)
  float* out = (float*)d_out;                      // [B*S, F]

  // Workspace carve-out (floats): UT 48MB | XT 32MB | outT 32MB | vals 1MB
  float* UT   = (float*)d_ws;
  float* XT   = UT   + (size_t)F_ * D_;
  float* outT = XT   + (size_t)F_ * BS_;
  float* vals = outT + (size_t)F_ * BS_;

  dim3 blk(32, 8);
  // up_decoder [768,16384] -> UT [16384,768]
  transpose_k<<<dim3(F_ / 32, D_ / 32), blk, 0, stream>>>(up_dec, UT, D_, F_);
  // up_facts [512,16384] -> XT [16384,512]
  transpose_k<<<dim3(F_ / 32, BS_ / 32), blk, 0, stream>>>(up_facts, XT, BS_, F_);
  // values via WMMA
  stage1_values<<<F_ / 16, 128, 0, stream>>>(down_enc, UT, jidx, vals);
  // gather-accumulate into outT [16384,512]
  stage2_scatter<<<F_, 256, 0, stream>>>(XT, vals, jidx, outT);
  // outT [16384,512] -> out [512,16384]
  transpose_k<<<dim3(BS_ / 32, F_ / 32), blk, 0, stream>>>(outT, out, F_, BS_);
}